// RGAT_69483980915102
// MI455X (gfx1250) — compile-verified
//
#include <hip/hip_runtime.h>
#include <hip/hip_bf16.h>
#include <math.h>

typedef float v2f __attribute__((ext_vector_type(2)));
typedef float v8f __attribute__((ext_vector_type(8)));

#define WAVE 32
#define DIM 128          // embedding dim (reference d)
#define F4_PER_ROW (DIM / 4)   // 32 float4 per row -> one per lane

__device__ __forceinline__ void atomic_add_f32(float* p, float v) {
    // native global_atomic_add_f32 (no CAS loop)
    unsafeAtomicAdd(p, v);
}

// ---------------------------------------------------------------------------
// WRT[r][i] = sum_k W[i][k] * rel[r][k]   (i in [0,2d), r in [0,R))
// i.e. WRT = (W @ rel^T)^T, stored row-major [R][2d] so that per-edge weight
// vectors are contiguous 1KB rows.
// One wave per 16x16 output tile, V_WMMA_F32_16X16X4_F32, K-loop over d.
// A[M][K] = W[m0+M][k],  B[K][N] = rel[n0+N][k],  D -> WRT[n0+N][m0+M]
// ---------------------------------------------------------------------------
__global__ void wr_gemm_kernel(const float* __restrict__ W,
                               const float* __restrict__ rel,
                               float* __restrict__ WRT,
                               int d, int R) {
    int wave = (blockIdx.x * blockDim.x + threadIdx.x) / WAVE;
    int lane = threadIdx.x & (WAVE - 1);
    int mtiles = (2 * d) / 16;              // 16 tiles over the 2d axis
    int ntiles = R / 16;                    // 4 tiles over relations
    if (wave >= mtiles * ntiles) return;
    int m0 = (wave % mtiles) * 16;
    int n0 = (wave / mtiles) * 16;

    int hi = lane >> 4;                     // 0: K pair {0,1}, 1: K pair {2,3}
    int lm = lane & 15;                     // M (for A) / N (for B)

    v8f c = {};
    const float* wrow = W + (size_t)(m0 + lm) * d;
    const float* rrow = rel + (size_t)(n0 + lm) * d;
    for (int k = 0; k < d; k += 4) {
        v2f a, b;
        a.x = wrow[k + 2 * hi];
        a.y = wrow[k + 2 * hi + 1];
        b.x = rrow[k + 2 * hi];
        b.y = rrow[k + 2 * hi + 1];
        c = __builtin_amdgcn_wmma_f32_16x16x4_f32(
                /*neg_a=*/false, a, /*neg_b=*/false, b,
                /*c_mod=*/(short)0, c, /*reuse_a=*/false, /*reuse_b=*/false);
    }
    // C/D layout: VGPR v, lanes 0-15 -> M=v, lanes 16-31 -> M=v+8; N=lane%16
    #pragma unroll
    for (int v = 0; v < 8; ++v) {
        int M = v + 8 * hi;
        WRT[(size_t)(n0 + lm) * (2 * d) + m0 + M] = c[v];
    }
}

// ---------------------------------------------------------------------------
// One wave per edge. Lane l holds float4 = elements [4l..4l+3] of a row.
// e = h_u . WRT[r][0:128] + t_e . WRT[r][128:256]; LeakyReLU(0.2); exp;
// atomically accumulate softmax denominators for head- and tail-grouping.
// ---------------------------------------------------------------------------
__global__ void edge_logits_kernel(const float* __restrict__ usr,
                                   const float* __restrict__ ent,
                                   const float* __restrict__ WRT,
                                   const int* __restrict__ eidx,
                                   const int* __restrict__ etype,
                                   float* __restrict__ e_exp,
                                   float* __restrict__ denom_u,
                                   float* __restrict__ denom_i,
                                   int E) {
    int wid  = (blockIdx.x * blockDim.x + threadIdx.x) >> 5;
    int lane = threadIdx.x & 31;
    if (wid >= E) return;
    int head = eidx[wid];
    int tail = eidx[E + wid];
    int r    = etype[wid];

    const float4* h4 = (const float4*)usr + (size_t)head * F4_PER_ROW;
    const float4* t4 = (const float4*)ent + (size_t)tail * F4_PER_ROW;
    const float4* v4 = (const float4*)WRT + (size_t)r * (2 * F4_PER_ROW);

    float4 hu = h4[lane];
    float4 te = t4[lane];
    float4 vt = v4[lane];                  // top half of W @ rel
    float4 vb = v4[F4_PER_ROW + lane];     // bottom half

    float s = hu.x * vt.x + hu.y * vt.y + hu.z * vt.z + hu.w * vt.w
            + te.x * vb.x + te.y * vb.y + te.z * vb.z + te.w * vb.w;
    #pragma unroll
    for (int off = 16; off; off >>= 1) s += __shfl_xor(s, off, 32);

    if (lane == 0) {
        float e  = s > 0.0f ? s : 0.2f * s;      // LeakyReLU(0.2)
        float ee = expf(e);
        e_exp[wid] = ee;
        atomic_add_f32(&denom_u[head], ee);
        atomic_add_f32(&denom_i[tail], ee);
    }
}

// ---------------------------------------------------------------------------
// One wave per edge: normalized attention weights, then weighted scatter-add
// of the 128-dim partner vector into agg_u[head] and agg_i[tail].
// ---------------------------------------------------------------------------
__global__ void edge_agg_kernel(const float* __restrict__ usr,
                                const float* __restrict__ ent,
                                const int* __restrict__ eidx,
                                const float* __restrict__ e_exp,
                                const float* __restrict__ denom_u,
                                const float* __restrict__ denom_i,
                                float* __restrict__ agg_u,
                                float* __restrict__ agg_i,
                                int E) {
    int wid  = (blockIdx.x * blockDim.x + threadIdx.x) >> 5;
    int lane = threadIdx.x & 31;
    if (wid >= E) return;
    int head = eidx[wid];
    int tail = eidx[E + wid];

    float ee = e_exp[wid];
    float eu = ee / denom_u[head];
    float ei = ee / denom_i[tail];

    float4 te = ((const float4*)ent)[(size_t)tail * F4_PER_ROW + lane];
    float4 hu = ((const float4*)usr)[(size_t)head * F4_PER_ROW + lane];

    float* pu = agg_u + (size_t)head * DIM + lane * 4;
    atomic_add_f32(pu + 0, te.x * eu);
    atomic_add_f32(pu + 1, te.y * eu);
    atomic_add_f32(pu + 2, te.z * eu);
    atomic_add_f32(pu + 3, te.w * eu);

    float* pi = agg_i + (size_t)tail * DIM + lane * 4;
    atomic_add_f32(pi + 0, hu.x * ei);
    atomic_add_f32(pi + 1, hu.y * ei);
    atomic_add_f32(pi + 2, hu.z * ei);
    atomic_add_f32(pi + 3, hu.w * ei);
}

// ---------------------------------------------------------------------------
// One wave per node: a = base + agg ; base += a / max(||a||, 1e-8)
// (agg holds only the segment-sum; the "+ base" residual is applied here)
// ---------------------------------------------------------------------------
__global__ void node_finalize_kernel(float* __restrict__ base,
                                     const float* __restrict__ agg,
                                     int N) {
    int wid  = (blockIdx.x * blockDim.x + threadIdx.x) >> 5;
    int lane = threadIdx.x & 31;
    if (wid >= N) return;

    float4 b = ((const float4*)base)[(size_t)wid * F4_PER_ROW + lane];
    float4 g = ((const float4*)agg)[(size_t)wid * F4_PER_ROW + lane];
    float4 a = make_float4(b.x + g.x, b.y + g.y, b.z + g.z, b.w + g.w);

    float ss = a.x * a.x + a.y * a.y + a.z * a.z + a.w * a.w;
    #pragma unroll
    for (int off = 16; off; off >>= 1) ss += __shfl_xor(ss, off, 32);

    float inv = 1.0f / fmaxf(sqrtf(ss), 1e-8f);
    float4 o = make_float4(b.x + a.x * inv, b.y + a.y * inv,
                           b.z + a.z * inv, b.w + a.w * inv);
    ((float4*)base)[(size_t)wid * F4_PER_ROW + lane] = o;
}

static inline size_t align256(size_t x) { return (x + 255) & ~(size_t)255; }

extern "C" void kernel_launch(void* const* d_in, const int* in_sizes, int n_in,
                              void* d_out, int out_size, void* d_ws, size_t ws_size,
                              hipStream_t stream) {
    const float* user_emb = (const float*)d_in[0];
    const float* rel_emb  = (const float*)d_in[1];
    const float* ent_emb  = (const float*)d_in[2];
    const float* W        = (const float*)d_in[3];
    const int*   eidx     = (const int*)d_in[4];
    const int*   etype    = (const int*)d_in[5];

    const int d  = DIM;
    const int Nu = in_sizes[0] / d;
    const int R  = in_sizes[1] / d;
    const int Ne = in_sizes[2] / d;
    const int E  = in_sizes[5];

    float* usr_all = (float*)d_out;                  // output slot 0
    float* ent_all = (float*)d_out + (size_t)Nu * d; // output slot 1

    // workspace carve-out
    char*  ws  = (char*)d_ws;
    size_t off = 0;
    float* WRT     = (float*)(ws + off); off = align256(off + (size_t)R * 2 * d * sizeof(float));
    float* e_exp   = (float*)(ws + off); off = align256(off + (size_t)E * sizeof(float));
    float* denom_u = (float*)(ws + off); off = align256(off + (size_t)Nu * sizeof(float));
    float* denom_i = (float*)(ws + off); off = align256(off + (size_t)Ne * sizeof(float));
    float* agg_u   = (float*)(ws + off); off = align256(off + (size_t)Nu * d * sizeof(float));
    float* agg_i   = (float*)(ws + off); off = align256(off + (size_t)Ne * d * sizeof(float));
    (void)ws_size; (void)n_in; (void)out_size;

    // seed outputs with the input embeddings (updated in place per hop)
    hipMemcpyAsync(usr_all, user_emb, (size_t)Nu * d * sizeof(float),
                   hipMemcpyDeviceToDevice, stream);
    hipMemcpyAsync(ent_all, ent_emb, (size_t)Ne * d * sizeof(float),
                   hipMemcpyDeviceToDevice, stream);

    // one-time tiny GEMM: WRT = (W @ rel^T)^T via f32 WMMA
    {
        int tiles   = ((2 * d) / 16) * (R / 16);     // 64 tiles
        int threads = tiles * WAVE;
        wr_gemm_kernel<<<(threads + 255) / 256, 256, 0, stream>>>(W, rel_emb, WRT, d, R);
    }

    const int blocksE = (int)(((size_t)E * WAVE + 255) / 256);
    for (int hop = 0; hop < 2; ++hop) {
        hipMemsetAsync(denom_u, 0, (size_t)Nu * sizeof(float), stream);
        hipMemsetAsync(denom_i, 0, (size_t)Ne * sizeof(float), stream);
        hipMemsetAsync(agg_u,   0, (size_t)Nu * d * sizeof(float), stream);
        hipMemsetAsync(agg_i,   0, (size_t)Ne * d * sizeof(float), stream);

        edge_logits_kernel<<<blocksE, 256, 0, stream>>>(
            usr_all, ent_all, WRT, eidx, etype, e_exp, denom_u, denom_i, E);

        edge_agg_kernel<<<blocksE, 256, 0, stream>>>(
            usr_all, ent_all, eidx, e_exp, denom_u, denom_i, agg_u, agg_i, E);

        node_finalize_kernel<<<(int)(((size_t)Nu * WAVE + 255) / 256), 256, 0, stream>>>(
            usr_all, agg_u, Nu);
        node_finalize_kernel<<<(int)(((size_t)Ne * WAVE + 255) / 256), 256, 0, stream>>>(
            ent_all, agg_i, Ne);
    }
}